// ExtractExclusivePatches_9285719294179
// MI455X (gfx1250) — compile-verified
//
#include <hip/hip_runtime.h>

// CDNA5 / gfx1250: wave32, WMMA 16x16x4 F32 path.
typedef float v2f __attribute__((ext_vector_type(2)));
typedef float v8f __attribute__((ext_vector_type(8)));

#define KK 9
#define CC 64

__device__ __forceinline__ float softplus_f(float x) {
  // numerically stable softplus: max(x,0) + log(1+exp(-|x|))
  float ax = fabsf(x);
  return fmaxf(x, 0.0f) + __logf(1.0f + __expf(-ax));
}

// One wave (32 lanes) per output segment. Events of segment s are the
// contiguous sorted range [start,end) in segment_ids_out. We accumulate the
// 9x64 patch as a 16x64 WMMA accumulator tile:
//   D[16 x 16ch_block] += A_onehot[16 x 4ev] * B_vals[4ev x 16ch]
// using v_wmma_f32_16x16x4_f32, 4 channel blocks, 4 events per step.
__global__ __launch_bounds__(128) void eep_gather_wmma_kernel(
    const float* __restrict__ feat,   // [E, 64]
    const float* __restrict__ dt,     // [E]
    const float* __restrict__ tout,   // [N]
    const int*   __restrict__ kid,    // [E]
    const int*   __restrict__ seg,    // [E] sorted
    const float* __restrict__ decay,  // [64]
    float*       __restrict__ out,    // [N, 9, 64]
    int E, int N) {
  const int lane = threadIdx.x & 31;
  const int wave = threadIdx.x >> 5;
  const int s = blockIdx.x * 4 + wave;
  if (s >= N) return;  // wave-uniform

  const int  m  = lane & 15;     // slot row (A) / channel column (B,D)
  const bool lo = lane < 16;     // lower half handles K=0,1; upper K=2,3

  // Per-lane decay rates for its 4 channel columns c = 16*b + m
  const float rate0 = softplus_f(decay[m]);
  const float rate1 = softplus_f(decay[16 + m]);
  const float rate2 = softplus_f(decay[32 + m]);
  const float rate3 = softplus_f(decay[48 + m]);

  // Binary search event range [start,end) for segment s (sorted seg[]).
  int start, end;
  {
    int a = 0, b = E;
    while (a < b) { int mid = (a + b) >> 1; if (seg[mid] < s) a = mid + 1; else b = mid; }
    start = a;
    b = E;
    while (a < b) { int mid = (a + b) >> 1; if (seg[mid] < s + 1) a = mid + 1; else b = mid; }
    end = a;
  }

  const float ts = tout[s];

  v8f acc0 = {}, acc1 = {}, acc2 = {}, acc3 = {};

  const int cnt    = end - start;
  const int nsteps = (cnt + 3) >> 2;
  for (int st = 0; st < nsteps; ++st) {
    const int base = start + (st << 2);
    // lane-half event pair: lower half -> events (base+0, base+1) = K 0/1
    //                       upper half -> events (base+2, base+3) = K 2/3
    int iA = base + (lo ? 0 : 2);
    int iB = base + (lo ? 1 : 3);
    const bool vA = iA < end;
    const bool vB = iB < end;
    iA = vA ? iA : (end - 1);  // clamp: contribution masked by A=0
    iB = vB ? iB : (end - 1);

    // gfx1250 prefetch of next step's feature rows (global_prefetch_b8)
    if (st + 1 < nsteps) {
      __builtin_prefetch(&feat[(size_t)(base + 4 + (lo ? 0 : 2)) * CC + m], 0, 1);
    }

    // A one-hot (slot row m gets 1 iff this event's kernel-id == m)
    const int kA = kid[iA];
    const int kB = kid[iB];
    v2f a;
    a.x = (vA && (kA == m)) ? 1.0f : 0.0f;
    a.y = (vB && (kB == m)) ? 1.0f : 0.0f;

    // decay weights; elapsed = ts - dt  =>  w = exp(rate * (dt - ts))
    const float negElA = dt[iA] - ts;
    const float negElB = dt[iB] - ts;
    const float wA0 = __expf(negElA * rate0), wB0 = __expf(negElB * rate0);
    const float wA1 = __expf(negElA * rate1), wB1 = __expf(negElB * rate1);
    const float wA2 = __expf(negElA * rate2), wB2 = __expf(negElB * rate2);
    const float wA3 = __expf(negElA * rate3), wB3 = __expf(negElB * rate3);

    const float* fA = feat + (size_t)iA * CC + m;
    const float* fB = feat + (size_t)iB * CC + m;

    v2f b0; b0.x = fA[0]  * wA0; b0.y = fB[0]  * wB0;
    v2f b1; b1.x = fA[16] * wA1; b1.y = fB[16] * wB1;
    v2f b2; b2.x = fA[32] * wA2; b2.y = fB[32] * wB2;
    v2f b3; b3.x = fA[48] * wA3; b3.y = fB[48] * wB3;

    acc0 = __builtin_amdgcn_wmma_f32_16x16x4_f32(false, a, false, b0, (short)0, acc0, false, false);
    acc1 = __builtin_amdgcn_wmma_f32_16x16x4_f32(false, a, false, b1, (short)0, acc1, false, false);
    acc2 = __builtin_amdgcn_wmma_f32_16x16x4_f32(false, a, false, b2, (short)0, acc2, false, false);
    acc3 = __builtin_amdgcn_wmma_f32_16x16x4_f32(false, a, false, b3, (short)0, acc3, false, false);
  }

  // Write the 9x64 patch. D layout (32-bit C/D 16x16): lanes 0-15 hold rows
  // 0-7 in vgprs 0-7 (col = lane), lanes 16-31 hold rows 8-15. We need rows
  // 0..8 only (K=9). Every slot written exactly once -> no zero-init pass.
  const size_t obase = (size_t)s * (KK * CC);
  {
    float* p0 = out + obase + 0 * 16 + m;
    float* p1 = out + obase + 1 * 16 + m;
    float* p2 = out + obase + 2 * 16 + m;
    float* p3 = out + obase + 3 * 16 + m;
    if (lo) {
#pragma unroll
      for (int r = 0; r < 8; ++r) {
        p0[(size_t)r * CC] = acc0[r];
        p1[(size_t)r * CC] = acc1[r];
        p2[(size_t)r * CC] = acc2[r];
        p3[(size_t)r * CC] = acc3[r];
      }
    } else {
      p0[(size_t)8 * CC] = acc0[0];
      p1[(size_t)8 * CC] = acc1[0];
      p2[(size_t)8 * CC] = acc2[0];
      p3[(size_t)8 * CC] = acc3[0];
    }
  }
}

extern "C" void kernel_launch(void* const* d_in, const int* in_sizes, int n_in,
                              void* d_out, int out_size, void* d_ws, size_t ws_size,
                              hipStream_t stream) {
  (void)n_in; (void)out_size; (void)d_ws; (void)ws_size;
  const float* feat  = (const float*)d_in[0];
  const float* dt    = (const float*)d_in[1];
  const float* tout  = (const float*)d_in[2];
  const int*   kid   = (const int*)d_in[3];
  const int*   seg   = (const int*)d_in[4];
  const float* decay = (const float*)d_in[5];
  float* out = (float*)d_out;

  const int E = in_sizes[1];  // number of input events (dt length)
  const int N = in_sizes[2];  // number of output segments (times_out length)

  const int segsPerBlock = 4;  // 4 waves of 32, one segment each
  const int grid = (N + segsPerBlock - 1) / segsPerBlock;
  eep_gather_wmma_kernel<<<grid, 128, 0, stream>>>(feat, dt, tout, kid, seg,
                                                   decay, out, E, N);
}